// UGFMEncoder_48464410968113
// MI455X (gfx1250) — compile-verified
//
#include <hip/hip_runtime.h>
#include <math.h>

// ---------------------------------------------------------------------------
// UGFM hetero-graph attention encoder for gfx1250 (MI455X).
//  * Edges binned by type (T), nodes by key (K) -> bucketed GEMMs via
//    V_WMMA_F32_16X16X4_F32 (fp32 tensor path, 16 k-steps over D=64).
//  * Weight matrices staged once per block in LDS (padded stride, conflict-
//    free b64 fragment loads), amortized over a grid-stride row-chunk loop.
//  * Edge softmax via float atomic-max / atomic-add in L2 (everything is
//    L2-resident on the 192MB L2 except the k/v streams).
//  * LayerNorm + readout: one wave32 per node, shfl_xor reductions.
// ---------------------------------------------------------------------------

typedef __attribute__((ext_vector_type(2))) float v2f;
typedef __attribute__((ext_vector_type(8))) float v8f;

#define WR 66   // padded LDS row stride (floats); even -> 8B-aligned b64 loads

__device__ __forceinline__ void atomicMaxF(float* addr, float v) {
  // monotone int-punning trick; init value is -INF
  if (v >= 0.0f) atomicMax((int*)addr, __float_as_int(v));
  else           atomicMin((unsigned int*)addr, __float_as_uint(v));
}

// ---------------- binning (labels are loop-invariant) ----------------------
__global__ void k_init0(int* kcnt, int K, int* tcnt, int T, float* osum) {
  int i = blockIdx.x * blockDim.x + threadIdx.x;
  if (i < K) kcnt[i] = 0;
  if (i < T) tcnt[i] = 0;
  if (i < 2) osum[i] = 0.0f;
}

__global__ void k_hist(const int* __restrict__ lab, int n, int* cnt) {
  int i = blockIdx.x * blockDim.x + threadIdx.x;
  if (i < n) atomicAdd(&cnt[lab[i]], 1);
}

__global__ void k_scan(const int* kcnt, int K, int* koff, int* kcur,
                       const int* tcnt, int T, int* toff, int* tcur) {
  if (threadIdx.x == 0 && blockIdx.x == 0) {
    int a = 0;
    for (int i = 0; i < K; ++i) { koff[i] = a; kcur[i] = a; a += kcnt[i]; }
    koff[K] = a;
    a = 0;
    for (int i = 0; i < T; ++i) { toff[i] = a; tcur[i] = a; a += tcnt[i]; }
    toff[T] = a;
  }
}

__global__ void k_scatter(const int* __restrict__ lab, int n, int* cur,
                          int* __restrict__ sorted) {
  int i = blockIdx.x * blockDim.x + threadIdx.x;
  if (i < n) {
    int p = atomicAdd(&cur[lab[i]], 1);
    sorted[p] = i;
  }
}

// ---------------- feat = emb[node_str] -------------------------------------
__global__ void k_gather_feat(const float* __restrict__ emb,
                              const int* __restrict__ nstr,
                              float* __restrict__ feat, int n64) {
  int i = blockIdx.x * blockDim.x + threadIdx.x;
  if (i < n64) {
    int node = i >> 6, d = i & 63;
    feat[i] = emb[(size_t)nstr[node] * 64 + d];
  }
}

// ---------------- per-conv state init --------------------------------------
__global__ void k_initconv(float* smax, float* den, float* agg, int N) {
  int i = blockIdx.x * blockDim.x + threadIdx.x;
  if (i < N * 64) agg[i] = 0.0f;
  if (i < N * 4) { smax[i] = -INFINITY; den[i] = 0.0f; }
}

// ---------------- bucketed GEMM via f32 WMMA -------------------------------
// Rows of bucket b are rowlist[boff[b]..boff[b+1]); A row = feat[srcmap?src:rid].
// C[row, hp] = sum_d A[row,d] * W[b][hp,d] + B[b][hp], for NMAT weight sets.
// Grid-stride loop over 128-row chunks so LDS weight staging is amortized.
template <int NMAT>
__global__ __launch_bounds__(256) void k_gemm_bucket(
    const float* __restrict__ feat,
    const int* __restrict__ rowlist,
    const int* __restrict__ boff,
    const int* __restrict__ srcmap,
    const float* __restrict__ W0, const float* __restrict__ B0,
    float* __restrict__ O0,
    const float* __restrict__ W1, const float* __restrict__ B1,
    float* __restrict__ O1) {
  __shared__ float sW[NMAT][64 * WR];
  __shared__ float sB[NMAT][64];

  const int b = blockIdx.x;
  const int r0 = boff[b];
  const int rows = boff[b + 1] - r0;
  if ((int)(blockIdx.y * 128) >= rows) return;   // uniform per block

  const int tid = threadIdx.x;
  // stage weight matrices (64x64 f32 each) + biases into LDS (padded stride)
  for (int mi = 0; mi < NMAT; ++mi) {
    const float* Wg = (mi == 0 ? W0 : W1) + (size_t)b * 4096;
    for (int i = tid; i < 1024; i += 256) {
      float4 w4 = ((const float4*)Wg)[i];
      int row = i >> 4, col = (i & 15) * 4;
      float* d = &sW[mi][row * WR + col];
      d[0] = w4.x; d[1] = w4.y; d[2] = w4.z; d[3] = w4.w;
    }
    if (tid < 64) sB[mi][tid] = (mi == 0 ? B0 : B1)[b * 64 + tid];
  }
  __syncthreads();

  const int wave = tid >> 5, lane = tid & 31;
  const int n = lane & 15, hi = lane >> 4;
  const int chunk_stride = gridDim.y * 128;      // rows per grid sweep

  for (int base = blockIdx.y * 128; base < rows; base += chunk_stride) {
    const int tile0 = base + wave * 16;
    if (tile0 >= rows) continue;                 // no barriers below: safe

    // A fragment source: this lane's matrix row (clamped; OOB masked at store)
    int ra = tile0 + n; if (ra > rows - 1) ra = rows - 1;
    int rid_a = rowlist[r0 + ra];
    int arow = srcmap ? srcmap[rid_a] : rid_a;
    const float* ap = feat + (size_t)arow * 64 + 2 * hi;

    v8f acc[NMAT][4];
    v8f zero = {0.f, 0.f, 0.f, 0.f, 0.f, 0.f, 0.f, 0.f};
    for (int mi = 0; mi < NMAT; ++mi)
      for (int nt = 0; nt < 4; ++nt) acc[mi][nt] = zero;

    for (int kc = 0; kc < 16; ++kc) {            // K = 64 in steps of 4
      v2f a = *(const v2f*)(ap + kc * 4);        // A[m][kc*4 + 2hi .. +1]
#pragma unroll
      for (int nt = 0; nt < 4; ++nt) {
#pragma unroll
        for (int mi = 0; mi < NMAT; ++mi) {
          v2f bf = *(const v2f*)&sW[mi][(nt * 16 + n) * WR + kc * 4 + 2 * hi];
          acc[mi][nt] = __builtin_amdgcn_wmma_f32_16x16x4_f32(
              false, a, false, bf, (short)0, acc[mi][nt], false, false);
        }
      }
    }

    // D layout: VGPR j -> row M = j + 8*hi, col N = nt*16 + n
    int erid[8];
#pragma unroll
    for (int j = 0; j < 8; ++j) {
      int row = tile0 + hi * 8 + j;
      erid[j] = (row < rows) ? rowlist[r0 + row] : -1;
    }
    for (int mi = 0; mi < NMAT; ++mi) {
      float* O = (mi == 0 ? O0 : O1);
#pragma unroll
      for (int nt = 0; nt < 4; ++nt) {
        float bias = sB[mi][nt * 16 + n];
#pragma unroll
        for (int j = 0; j < 8; ++j) {
          if (erid[j] >= 0)
            O[(size_t)erid[j] * 64 + nt * 16 + n] = acc[mi][nt][j] + bias;
        }
      }
    }
  }
}

// ---------------- edge softmax pipeline ------------------------------------
__global__ void k_score(const int* __restrict__ edst,
                        const float* __restrict__ q,
                        const float* __restrict__ kb,
                        float* __restrict__ score,
                        float* __restrict__ smax, int EH) {
  int i = blockIdx.x * blockDim.x + threadIdx.x;
  if (i >= EH) return;
  int e = i >> 2, h = i & 3;
  int dst = edst[e];
  const float* qp = q + (size_t)dst * 64 + h * 16;
  const float* kp = kb + (size_t)e * 64 + h * 16;
  float s = 0.0f;
#pragma unroll
  for (int p = 0; p < 16; ++p) s += qp[p] * kp[p];
  s *= 0.25f;                                  // 1/sqrt(P), P=16
  score[i] = s;
  atomicMaxF(&smax[dst * 4 + h], s);
}

__global__ void k_exden(const int* __restrict__ edst,
                        float* __restrict__ score,      // in: score, out: exp
                        const float* __restrict__ smax,
                        float* __restrict__ den, int EH) {
  int i = blockIdx.x * blockDim.x + threadIdx.x;
  if (i >= EH) return;
  int e = i >> 2, h = i & 3;
  int dst = edst[e];
  float mx = smax[dst * 4 + h];
  if (!(mx > -1e37f)) mx = 0.0f;               // reference: non-finite -> 0
  float ex = __expf(score[i] - mx);
  score[i] = ex;
  atomicAdd(&den[dst * 4 + h], ex);
}

__global__ void k_agg(const int* __restrict__ edst,
                      const float* __restrict__ ex,
                      const float* __restrict__ den,
                      const float* __restrict__ vb,
                      float* __restrict__ agg, int EH) {
  int i = blockIdx.x * blockDim.x + threadIdx.x;
  if (i >= EH) return;
  int e = i >> 2, h = i & 3;
  int dst = edst[e];
  float a = ex[i] / (den[dst * 4 + h] + 1e-9f);
  const float* vp = vb + (size_t)e * 64 + h * 16;
  float* ap = agg + (size_t)dst * 64 + h * 16;
#pragma unroll
  for (int p = 0; p < 16; ++p) atomicAdd(&ap[p], a * vp[p]);
}

// ---------------- ReLU + LayerNorm + readout (1 wave32 per node) -----------
__global__ void k_update(const float* __restrict__ agg,
                         float* __restrict__ feat,
                         const int* __restrict__ nkey,
                         const float* __restrict__ tw,
                         float* __restrict__ osum, int conv, int N) {
  int gi = blockIdx.x * blockDim.x + threadIdx.x;
  int node = gi >> 5, lane = gi & 31;
  if (node >= N) return;
  float x0 = fmaxf(agg[(size_t)node * 64 + lane], 0.0f);
  float x1 = fmaxf(agg[(size_t)node * 64 + lane + 32], 0.0f);
  float s = x0 + x1, s2 = x0 * x0 + x1 * x1;
#pragma unroll
  for (int o = 16; o > 0; o >>= 1) {
    s  += __shfl_xor(s, o, 32);
    s2 += __shfl_xor(s2, o, 32);
  }
  float m   = s * (1.0f / 64.0f);
  float var = s2 * (1.0f / 64.0f) - m * m;
  float inv = rsqrtf(var + 1e-5f);
  float y0 = (x0 - m) * inv, y1 = (x1 - m) * inv;
  feat[(size_t)node * 64 + lane]      = y0;
  feat[(size_t)node * 64 + lane + 32] = y1;
  int key = nkey[node];
  float t = y0 * tw[(size_t)key * 64 + lane] + y1 * tw[(size_t)key * 64 + lane + 32];
#pragma unroll
  for (int o = 16; o > 0; o >>= 1) t += __shfl_xor(t, o, 32);
  if (lane == 0) atomicAdd(&osum[conv], t);
}

__global__ void k_finalize(const float* osum, float* out, int N) {
  if (threadIdx.x == 0 && blockIdx.x == 0)
    out[0] = (osum[0] + osum[1]) * (0.5f / (float)N);
}

// ---------------------------------------------------------------------------
extern "C" void kernel_launch(void* const* d_in, const int* in_sizes, int n_in,
                              void* d_out, int out_size, void* d_ws, size_t ws_size,
                              hipStream_t stream) {
  const float* emb   = (const float*)d_in[0];
  const int*   nstr  = (const int*)d_in[1];
  const int*   nkey  = (const int*)d_in[2];
  const int*   esrc  = (const int*)d_in[3];
  const int*   edst  = (const int*)d_in[4];
  const int*   etyp  = (const int*)d_in[5];
  const float* qw    = (const float*)d_in[6];
  const float* qb    = (const float*)d_in[7];
  const float* kw    = (const float*)d_in[8];
  const float* kbia  = (const float*)d_in[9];
  const float* vw    = (const float*)d_in[10];
  const float* vbia  = (const float*)d_in[11];
  const float* tw    = (const float*)d_in[12];
  float* out = (float*)d_out;

  const int N = in_sizes[2];
  const int E = in_sizes[3];
  const int K = in_sizes[7] / 64;   // query_b = [1,K,H,P], H*P = 64
  const int T = in_sizes[9] / 64;   // key_b   = [1,T,H,P]

  // ---- workspace carve (bump allocator, 256B aligned) ----
  char* p = (char*)d_ws;
  auto carve = [&](size_t bytes) -> char* {
    char* r = p;
    p += (bytes + 255) & ~(size_t)255;
    return r;
  };
  float* feat  = (float*)carve((size_t)N * 64 * 4);
  float* q     = (float*)carve((size_t)N * 64 * 4);
  float* kb    = (float*)carve((size_t)E * 64 * 4);
  float* vb    = (float*)carve((size_t)E * 64 * 4);
  float* score = (float*)carve((size_t)E * 4 * 4);
  float* smax  = (float*)carve((size_t)N * 4 * 4);
  float* den   = (float*)carve((size_t)N * 4 * 4);
  float* agg   = (float*)carve((size_t)N * 64 * 4);
  int* kcnt    = (int*)carve((size_t)K * 4);
  int* koff    = (int*)carve((size_t)(K + 1) * 4);
  int* kcur    = (int*)carve((size_t)K * 4);
  int* nsort   = (int*)carve((size_t)N * 4);
  int* tcnt    = (int*)carve((size_t)T * 4);
  int* toff    = (int*)carve((size_t)(T + 1) * 4);
  int* tcur    = (int*)carve((size_t)T * 4);
  int* esort   = (int*)carve((size_t)E * 4);
  float* osum  = (float*)carve(2 * 4);

  const int B = 256;
  auto g = [&](long long n) { return (unsigned)((n + B - 1) / B); };
  auto mini = [](unsigned a, unsigned b) { return a < b ? a : b; };

  // ---- one-time binning + embedding gather ----
  k_init0<<<1, 128, 0, stream>>>(kcnt, K, tcnt, T, osum);
  k_hist<<<g(N), B, 0, stream>>>(nkey, N, kcnt);
  k_hist<<<g(E), B, 0, stream>>>(etyp, E, tcnt);
  k_scan<<<1, 32, 0, stream>>>(kcnt, K, koff, kcur, tcnt, T, toff, tcur);
  k_scatter<<<g(N), B, 0, stream>>>(nkey, N, kcur, nsort);
  k_scatter<<<g(E), B, 0, stream>>>(etyp, E, tcur, esort);
  k_gather_feat<<<g((long long)N * 64), B, 0, stream>>>(emb, nstr, feat, N * 64);

  // GEMM grids: x = bucket, y = row-chunk sweep (grid-stride inside kernel).
  // y capped so weight staging (33KB/block) is amortized over ~20+ chunks,
  // while still launching 640-1280 workgroups to fill the WGPs.
  const unsigned yq = mini(10u, (unsigned)((N + 127) / 128));
  const unsigned ye = mini(40u, (unsigned)((E + 127) / 128));

  // ---- two convolutions ----
  for (int conv = 0; conv < 2; ++conv) {
    k_initconv<<<g((long long)N * 64), B, 0, stream>>>(smax, den, agg, N);

    // q = feat . Wq[key]^T + bq      (node buckets, single matrix)
    dim3 gq((unsigned)K, yq, 1);
    k_gemm_bucket<1><<<gq, 256, 0, stream>>>(feat, nsort, koff, nullptr,
                                             qw, qb, q,
                                             nullptr, nullptr, nullptr);
    // k,v = feat[src] . W{k,v}[type]^T + b   (edge buckets, dual matrices)
    dim3 ge((unsigned)T, ye, 1);
    k_gemm_bucket<2><<<ge, 256, 0, stream>>>(feat, esort, toff, esrc,
                                             kw, kbia, kb,
                                             vw, vbia, vb);

    k_score<<<g((long long)E * 4), B, 0, stream>>>(edst, q, kb, score, smax, E * 4);
    k_exden<<<g((long long)E * 4), B, 0, stream>>>(edst, score, smax, den, E * 4);
    k_agg<<<g((long long)E * 4), B, 0, stream>>>(edst, score, den, vb, agg, E * 4);
    k_update<<<g((long long)N * 32), B, 0, stream>>>(agg, feat, nkey, tw, osum, conv, N);
  }

  k_finalize<<<1, 32, 0, stream>>>(osum, out, N);
}